// GatedGCNConvLayer_65455301591165
// MI455X (gfx1250) — compile-verified
//
#include <hip/hip_runtime.h>

typedef float v2f __attribute__((ext_vector_type(2)));
typedef float v4f __attribute__((ext_vector_type(4)));
typedef float v8f __attribute__((ext_vector_type(8)));

#define DIM 64

// ---------------------------------------------------------------------------
// Edge scatter (defined FIRST so the disasm snippet shows its atomic
// lowering). Each wave handles 2 edges: lanes 0-15 -> edge e0, lanes 16-31 ->
// edge e1; each lane covers 4 dims via b128 loads (16 lanes x 16B = one
// coalesced 256B feature row per edge). k/q/v (76.8MB) are L2-resident on the
// 192MB L2, so gathers are L2 hits. Messages are accumulated with native fp32
// atomics (relaxed, agent scope) into d_out, which already holds
// x@W_skip + b_skip from the projection kernel (stream-ordered).
// ---------------------------------------------------------------------------
__global__ __launch_bounds__(256)
void edge_gate_scatter_kernel(const long long* __restrict__ edge_index,
                              const float* __restrict__ k,
                              const float* __restrict__ q,
                              const float* __restrict__ v,
                              float* __restrict__ out,
                              long long E)
{
    const int wid  = threadIdx.x >> 5;           // wave in block (wave32)
    const int lane = threadIdx.x & 31;
    const int half = lane >> 4;                  // which edge within the wave
    const int sub  = lane & 15;                  // lane within the half

    const long long e = ((long long)blockIdx.x * 8 + wid) * 2 + half;
    if (e >= E) return;                          // half-wave-uniform guard

    const long long s = edge_index[e];           // source j
    const long long d = edge_index[E + e];       // target i

    const int c = sub * 4;                       // 4 dims per lane
    const v4f kk = *(const v4f*)(k + d * DIM + c);
    const v4f qq = *(const v4f*)(q + s * DIM + c);
    const v4f vv = *(const v4f*)(v + s * DIM + c);

    float* o = out + d * DIM + c;
    #pragma unroll
    for (int i = 0; i < 4; ++i) {
        const float g = 1.0f / (1.0f + __expf(-(kk[i] + qq[i])));
        __hip_atomic_fetch_add(o + i, g * vv[i],
                               __ATOMIC_RELAXED, __HIP_MEMORY_SCOPE_AGENT);
    }
}

// ---------------------------------------------------------------------------
// Fused 4-way projection GEMM using V_WMMA_F32_16X16X4_F32 (fp32 matrix pipe,
// full reference precision). Each wave computes a 16-row stripe of ONE
// projection matrix (key / query / value / skip) across all 64 columns:
// 4 n-tile accumulators (v8f each) x 16 K-steps = 64 WMMAs per wave.
// A-fragment layout (16x4 f32): lane<16 -> M=lane, K={k0,k0+1};
// lane>=16 -> M=lane-16, K={k0+2,k0+3}. B (4x16) mirrors it with N=lane&15.
// C/D: VGPR r -> rows r / r+8, N = lane&15.
// ---------------------------------------------------------------------------
__global__ __launch_bounds__(256)
void proj4_wmma_kernel(const float* __restrict__ x,
                       const float* __restrict__ Wk, const float* __restrict__ bk,
                       const float* __restrict__ Wq, const float* __restrict__ bq,
                       const float* __restrict__ Wv, const float* __restrict__ bv,
                       const float* __restrict__ Ws, const float* __restrict__ bs,
                       float* __restrict__ k_out, float* __restrict__ q_out,
                       float* __restrict__ v_out, float* __restrict__ skip_out,
                       int N)
{
    const int wid  = threadIdx.x >> 5;
    const int lane = threadIdx.x & 31;
    const long long gw = (long long)blockIdx.x * 8 + wid;

    const int mat      = (int)(gw & 3);
    const long long rt = gw >> 2;                // 16-row tile index
    const long long m0 = rt * 16;
    if (m0 >= N) return;                         // wave-uniform exit

    const int half = lane >> 4;
    const int lrow = lane & 15;

    const float* W = (mat == 0) ? Wk : (mat == 1) ? Wq : (mat == 2) ? Wv : Ws;
    const float* B = (mat == 0) ? bk : (mat == 1) ? bq : (mat == 2) ? bv : bs;
    float* dst     = (mat == 0) ? k_out : (mat == 1) ? q_out : (mat == 2) ? v_out : skip_out;

    long long arow = m0 + lrow;                  // defensive clamp (N%16==0)
    if (arow >= N) arow = N - 1;
    const float* xrow = x + arow * DIM;

    v8f acc0 = {}, acc1 = {}, acc2 = {}, acc3 = {};

    #pragma unroll
    for (int kk = 0; kk < 16; ++kk) {
        const int k0 = kk * 4;
        const int ks = k0 + 2 * half;            // this lane's K pair
        v2f a = *(const v2f*)(xrow + ks);        // A fragment (b64 load)

        #pragma unroll
        for (int nt = 0; nt < 4; ++nt) {
            const int n = nt * 16 + lrow;
            v2f b;
            b.x = W[(long long)ks * DIM + n];
            b.y = W[(long long)(ks + 1) * DIM + n];
            switch (nt) {
            case 0: acc0 = __builtin_amdgcn_wmma_f32_16x16x4_f32(false, a, false, b, (short)0, acc0, false, false); break;
            case 1: acc1 = __builtin_amdgcn_wmma_f32_16x16x4_f32(false, a, false, b, (short)0, acc1, false, false); break;
            case 2: acc2 = __builtin_amdgcn_wmma_f32_16x16x4_f32(false, a, false, b, (short)0, acc2, false, false); break;
            case 3: acc3 = __builtin_amdgcn_wmma_f32_16x16x4_f32(false, a, false, b, (short)0, acc3, false, false); break;
            }
        }
    }

    #pragma unroll
    for (int nt = 0; nt < 4; ++nt) {
        const int n = nt * 16 + lrow;
        const float bias = B[n];
        const v8f acc = (nt == 0) ? acc0 : (nt == 1) ? acc1 : (nt == 2) ? acc2 : acc3;
        #pragma unroll
        for (int r = 0; r < 8; ++r) {
            const long long row = m0 + r + 8 * half;
            if (row < N) dst[row * DIM + n] = acc[r] + bias;
        }
    }
}

extern "C" void kernel_launch(void* const* d_in, const int* in_sizes, int n_in,
                              void* d_out, int out_size, void* d_ws, size_t ws_size,
                              hipStream_t stream)
{
    const float*      x  = (const float*)d_in[0];
    const long long*  ei = (const long long*)d_in[1];
    const float* Wk = (const float*)d_in[2];
    const float* bk = (const float*)d_in[3];
    const float* Wq = (const float*)d_in[4];
    const float* bq = (const float*)d_in[5];
    const float* Wv = (const float*)d_in[6];
    const float* bv = (const float*)d_in[7];
    const float* Ws = (const float*)d_in[8];
    const float* bs = (const float*)d_in[9];
    float* out = (float*)d_out;

    const int       N = in_sizes[0] / DIM;          // 100000
    const long long E = (long long)in_sizes[1] / 2; // 1600000

    float* k = (float*)d_ws;
    float* q = k + (size_t)N * DIM;
    float* v = q + (size_t)N * DIM;

    // Projections: 4 matrices x ceil(N/16) row tiles, 1 wave each, 8 waves/block.
    const long long row_tiles = (N + 15) / 16;
    const long long waves     = row_tiles * 4;
    const int proj_blocks     = (int)((waves + 7) / 8);
    proj4_wmma_kernel<<<proj_blocks, 256, 0, stream>>>(
        x, Wk, bk, Wq, bq, Wv, bv, Ws, bs, k, q, v, out, N);

    // Edge scatter: 2 edges per wave, 8 waves per block -> 16 edges/block.
    const int edge_blocks = (int)((E + 15) / 16);
    edge_gate_scatter_kernel<<<edge_blocks, 256, 0, stream>>>(ei, k, q, v, out, E);
}